// GeneSAGE_71373766525394
// MI455X (gfx1250) — compile-verified
//
#include <hip/hip_runtime.h>
#include <hip/hip_bf16.h>

#define N_NODES 50000
#define N_EDGES 800000
#define IN_DIM 64
#define HID 256
#define OUT_DIM 2
#define LN_EPS 1e-5f

typedef __attribute__((ext_vector_type(2))) float v2f;
typedef __attribute__((ext_vector_type(8))) float v8f;

// ---------------------------------------------------------------- zero scratch
__global__ void gs_zero(float* __restrict__ p, long long n) {
    long long i = (long long)blockIdx.x * blockDim.x + threadIdx.x;
    long long stride = (long long)gridDim.x * blockDim.x;
    for (; i < n; i += stride) p[i] = 0.0f;
}

// ------------------------------------------------- fold W1r+Wskip, b1+bskip
__global__ void gs_prep(const float* __restrict__ W1r, const float* __restrict__ Wskip,
                        const float* __restrict__ b1, const float* __restrict__ bskip,
                        float* __restrict__ Wcomb, float* __restrict__ bcomb) {
    int i = blockIdx.x * blockDim.x + threadIdx.x;
    if (i < IN_DIM * HID) Wcomb[i] = W1r[i] + Wskip[i];
    if (i < HID) bcomb[i] = b1[i] + bskip[i];
}

// ------------------------------------- layer-1 edge scatter: agg1 += x[src], cnt
__global__ void gs_agg1(const int* __restrict__ ei, const float* __restrict__ x,
                        float* __restrict__ agg1, float* __restrict__ cnt) {
    unsigned long long gid = (unsigned long long)blockIdx.x * blockDim.x + threadIdx.x;
    int e = (int)(gid >> 6);
    int d = (int)(gid & 63);
    if (e >= N_EDGES) return;
    int src = ei[e];
    int dst = ei[N_EDGES + e];
    atomicAdd(&agg1[(long long)dst * IN_DIM + d], x[(long long)src * IN_DIM + d]);
    if (d == 0) atomicAdd(&cnt[dst], 1.0f);
}

// ---- layer-1 GEMM via V_WMMA_F32_16X16X4_F32, fused bias + LayerNorm + ELU ----
// block: 256 threads (8 waves). Handles 16 node rows x 256 output cols.
// Each wave computes two 16x16 output tiles: C += meanA * W1l + xA * Wcomb.
__global__ void gs_gemm1_ln_elu(const float* __restrict__ x, const float* __restrict__ agg1,
                                const float* __restrict__ cnt, const float* __restrict__ W1l,
                                const float* __restrict__ Wcomb, const float* __restrict__ bcomb,
                                const float* __restrict__ gamma, const float* __restrict__ beta,
                                float* __restrict__ h) {
    __shared__ float xs[16][HID + 4];

    const int m0 = blockIdx.x * 16;
    const int wave = threadIdx.x >> 5;
    const int lane = threadIdx.x & 31;
    const int half = lane >> 4;    // 0: lanes 0-15, 1: lanes 16-31
    const int l16  = lane & 15;

    const int  arow = m0 + l16;                      // A-fragment node row
    const float inv = 1.0f / fmaxf(cnt[arow], 1.0f); // mean divisor

    #pragma unroll
    for (int t = 0; t < 2; ++t) {
        const int n0 = (wave * 2 + t) * 16;
        v8f c = {};
        #pragma unroll
        for (int k0 = 0; k0 < IN_DIM; k0 += 4) {
            const int kb = k0 + half * 2;  // ISA f32 A layout: VGPR0/1 = K{0,1} | K{2,3}
            v2f am, ax, bl, bc;
            am.x = agg1[(long long)arow * IN_DIM + kb]     * inv;
            am.y = agg1[(long long)arow * IN_DIM + kb + 1] * inv;
            ax.x = x[(long long)arow * IN_DIM + kb];
            ax.y = x[(long long)arow * IN_DIM + kb + 1];
            bl.x = W1l[(kb)     * HID + n0 + l16];
            bl.y = W1l[(kb + 1) * HID + n0 + l16];
            bc.x = Wcomb[(kb)     * HID + n0 + l16];
            bc.y = Wcomb[(kb + 1) * HID + n0 + l16];
            c = __builtin_amdgcn_wmma_f32_16x16x4_f32(false, am, false, bl, (short)0, c, false, false);
            c = __builtin_amdgcn_wmma_f32_16x16x4_f32(false, ax, false, bc, (short)0, c, false, false);
        }
        // C/D layout: VGPR v -> row v (lanes 0-15) / row v+8 (lanes 16-31)
        const float bb = bcomb[n0 + l16];
        #pragma unroll
        for (int v = 0; v < 8; ++v)
            xs[v + half * 8][n0 + l16] = c[v] + bb;
    }
    __syncthreads();

    // LayerNorm + ELU: 16-thread team per row; teams align with half-waves.
    const int row = threadIdx.x >> 4;  // 0..15
    const int j   = threadIdx.x & 15;
    float s = 0.0f, ss = 0.0f;
    #pragma unroll
    for (int i = 0; i < 16; ++i) {
        float v = xs[row][j + 16 * i];
        s += v; ss += v * v;
    }
    #pragma unroll
    for (int m = 1; m < 16; m <<= 1) {      // stays within each 16-lane half
        s  += __shfl_xor(s,  m, 32);
        ss += __shfl_xor(ss, m, 32);
    }
    const float mu  = s * (1.0f / HID);
    const float var = ss * (1.0f / HID) - mu * mu;
    const float rs  = rsqrtf(var + LN_EPS);
    const long long node = m0 + row;
    #pragma unroll
    for (int i = 0; i < 16; ++i) {
        int col = j + 16 * i;
        float v = (xs[row][col] - mu) * rs * gamma[col] + beta[col];
        h[node * HID + col] = v > 0.0f ? v : expm1f(v);   // ELU(alpha=1)
    }
}

// ------------------- layer-2 edge scatter: agg2[dst] += h[src] (block per edge)
__global__ void gs_agg2(const int* __restrict__ ei, const float* __restrict__ hbuf,
                        float* __restrict__ agg2) {
    const int e = blockIdx.x;
    const int d = threadIdx.x;  // 256 threads = HID
    const int src = ei[e];
    const int dst = ei[N_EDGES + e];
    atomicAdd(&agg2[(long long)dst * HID + d], hbuf[(long long)src * HID + d]);
}

// ------------ layer-2: out = (agg2/cnt)@W2l + h@W2r + b2, one wave per node
__global__ void gs_out(const float* __restrict__ hbuf, const float* __restrict__ agg2,
                       const float* __restrict__ cnt, const float* __restrict__ W2l,
                       const float* __restrict__ W2r, const float* __restrict__ b2,
                       float* __restrict__ out) {
    const int wave = threadIdx.x >> 5;
    const int lane = threadIdx.x & 31;
    const int n = blockIdx.x * 8 + wave;
    if (n >= N_NODES) return;
    const float inv = 1.0f / fmaxf(cnt[n], 1.0f);
    float a0 = 0.0f, a1 = 0.0f;
    #pragma unroll
    for (int i = 0; i < 8; ++i) {
        int k = lane + 32 * i;
        float hv = hbuf[(long long)n * HID + k];
        float mv = agg2[(long long)n * HID + k] * inv;
        a0 += mv * W2l[k * 2 + 0] + hv * W2r[k * 2 + 0];
        a1 += mv * W2l[k * 2 + 1] + hv * W2r[k * 2 + 1];
    }
    #pragma unroll
    for (int m = 1; m < 32; m <<= 1) {
        a0 += __shfl_xor(a0, m, 32);
        a1 += __shfl_xor(a1, m, 32);
    }
    if (lane == 0) {
        out[n * 2 + 0] = a0 + b2[0];
        out[n * 2 + 1] = a1 + b2[1];
    }
}

extern "C" void kernel_launch(void* const* d_in, const int* in_sizes, int n_in,
                              void* d_out, int out_size, void* d_ws, size_t ws_size,
                              hipStream_t stream) {
    const float* x     = (const float*)d_in[0];
    const int*   ei    = (const int*)d_in[1];
    const float* W1l   = (const float*)d_in[2];
    const float* W1r   = (const float*)d_in[3];
    const float* b1    = (const float*)d_in[4];
    const float* Wskip = (const float*)d_in[5];
    const float* bskip = (const float*)d_in[6];
    const float* gamma = (const float*)d_in[7];
    const float* beta  = (const float*)d_in[8];
    const float* W2l   = (const float*)d_in[9];
    const float* W2r   = (const float*)d_in[10];
    const float* b2    = (const float*)d_in[11];
    float* out = (float*)d_out;

    // workspace layout (floats)
    float* ws   = (float*)d_ws;
    float* agg1 = ws;                                    // N*64
    float* cnt  = agg1 + (long long)N_NODES * IN_DIM;    // N
    float* agg2 = cnt + N_NODES;                         // N*256
    float* hbuf = agg2 + (long long)N_NODES * HID;       // N*256
    float* Wcomb = hbuf + (long long)N_NODES * HID;      // 64*256
    float* bcomb = Wcomb + IN_DIM * HID;                 // 256

    const long long zero_n = (long long)N_NODES * IN_DIM + N_NODES + (long long)N_NODES * HID;
    gs_zero<<<4096, 256, 0, stream>>>(ws, zero_n);
    gs_prep<<<(IN_DIM * HID + 255) / 256, 256, 0, stream>>>(W1r, Wskip, b1, bskip, Wcomb, bcomb);

    const long long a1_threads = (long long)N_EDGES * IN_DIM;
    gs_agg1<<<(unsigned)((a1_threads + 255) / 256), 256, 0, stream>>>(ei, x, agg1, cnt);

    gs_gemm1_ln_elu<<<N_NODES / 16, 256, 0, stream>>>(x, agg1, cnt, W1l, Wcomb, bcomb,
                                                      gamma, beta, hbuf);

    gs_agg2<<<N_EDGES, HID, 0, stream>>>(ei, hbuf, agg2);

    gs_out<<<N_NODES / 8, 256, 0, stream>>>(hbuf, agg2, cnt, W2l, W2r, b2, out);
}